// myNN_57183194579644
// MI455X (gfx1250) — compile-verified
//
#include <hip/hip_runtime.h>
#include <hip/hip_bf16.h>

#define VOCAB 256

// Native 16-byte vector (clang ext_vector): valid for __builtin_nontemporal_store
// and lowers to b128 loads/stores.
typedef float v4f __attribute__((ext_vector_type(4)));

// ---------------------------------------------------------------------------
// Kernel 1: build fused lookup table T[v][o] = W[o][v] + b[o]  (256 KB)
// One-time transpose+bias so each token's output row is a contiguous 1 KB
// block. Tiny (64K elements); launched with 256 blocks x 256 threads.
// ---------------------------------------------------------------------------
__global__ void build_table_kernel(const float* __restrict__ W,
                                   const float* __restrict__ b,
                                   float* __restrict__ T) {
    const int v = blockIdx.x;    // input vocab index (column of W)
    const int o = threadIdx.x;   // output channel   (row of W)
    T[v * VOCAB + o] = W[o * VOCAB + v] + b[o];
}

// ---------------------------------------------------------------------------
// Kernel 2: bandwidth-bound gather. Stage the whole 256 KB table into LDS
// with the CDNA5 async global->LDS copy engine, then stream output rows.
//   out[t*256 + o] = T[x[t]*256 + o]
// ---------------------------------------------------------------------------
__global__ void __launch_bounds__(1024)
onehot_linear_gather_kernel(const int* __restrict__ x,
                            const float* __restrict__ T,
                            float* __restrict__ out,
                            int ntok) {
    // Full table in LDS: 256*256 f32 = 256 KB (CDNA5 WGP has 320 KB LDS).
    __shared__ float tab[VOCAB * VOCAB];

    const int tid = threadIdx.x;  // 0..1023

    // ---- Stage T -> LDS with async b128 copies (ASYNCcnt-tracked) --------
    // 16384 16B chunks total; each thread issues 16 fully-coalesced
    // global_load_async_to_lds_b128 ops (consecutive tids -> consecutive
    // 16B chunks -> 512 B per wave per issue).
    const v4f* __restrict__ Tv = (const v4f*)T;
    #pragma unroll
    for (int i = 0; i < 16; ++i) {
        const int c4 = tid + i * 1024;                      // 16B-chunk index
        unsigned lds_addr =
            (unsigned)(size_t)(&tab[c4 * 4]);               // LDS byte offset
        unsigned long long gaddr =
            (unsigned long long)(size_t)(Tv + c4);          // global byte addr
        asm volatile("global_load_async_to_lds_b128 %0, %1, off"
                     :: "v"(lds_addr), "v"(gaddr)
                     : "memory");
    }
    // Each wave waits for its own async copies, then the workgroup barrier
    // makes the full table visible to all 32 waves.
    asm volatile("s_wait_asynccnt 0x0" ::: "memory");
    __syncthreads();

    // ---- Stream output rows ---------------------------------------------
    // Contiguous token chunk per wave: the wave's stores form one linear
    // multi-KB stream. Per token, lane l covers floats [8l, 8l+8):
    // 2x ds_load_b128 from LDS, 2x non-temporal global_store_b128
    // (don't pollute L2 with the 268 MB streaming output).
    const int lane   = tid & 31;
    const int wave   = tid >> 5;
    const int gwave  = blockIdx.x * (blockDim.x >> 5) + wave;
    const int nwaves = gridDim.x * (blockDim.x >> 5);
    const int chunk  = (ntok + nwaves - 1) / nwaves;
    const int t0     = gwave * chunk;
    const int t1     = (t0 + chunk < ntok) ? (t0 + chunk) : ntok;

    const v4f* __restrict__ ltab = (const v4f*)tab;

    for (int t = t0; t < t1; ++t) {
        const int v = x[t];                      // uniform across the wave
        const v4f r0 = ltab[v * 64 + lane * 2 + 0];
        const v4f r1 = ltab[v * 64 + lane * 2 + 1];
        v4f* __restrict__ dst = ((v4f*)out) + (size_t)t * 64 + lane * 2;
        __builtin_nontemporal_store(r0, dst + 0);
        __builtin_nontemporal_store(r1, dst + 1);
    }
}

// ---------------------------------------------------------------------------
// Host-side launch. Inputs (setup_inputs order): x [32*8192] int, W [256*256]
// f32, b [256] f32. Output: [32*8192*256] f32. d_ws holds the 256 KB table.
// ---------------------------------------------------------------------------
extern "C" void kernel_launch(void* const* d_in, const int* in_sizes, int n_in,
                              void* d_out, int out_size, void* d_ws, size_t ws_size,
                              hipStream_t stream) {
    const int*   x = (const int*)d_in[0];
    const float* W = (const float*)d_in[1];
    const float* b = (const float*)d_in[2];
    float* out = (float*)d_out;
    float* T   = (float*)d_ws;          // 256 KB scratch table
    const int ntok = in_sizes[0];       // 32 * 8192 = 262144

    // One 256KB-LDS block per compute unit (each WGP fits exactly one).
    // Attribute queries are deterministic and legal during graph capture.
    int dev = 0, ncu = 0;
    (void)hipGetDevice(&dev);
    if (hipDeviceGetAttribute(&ncu, hipDeviceAttributeMultiprocessorCount,
                              dev) != hipSuccess || ncu <= 0) {
        ncu = 512;
    }
    int blocks = ncu;
    if (blocks < 256)  blocks = 256;
    if (blocks > 2048) blocks = 2048;

    build_table_kernel<<<VOCAB, VOCAB, 0, stream>>>(W, b, T);
    onehot_linear_gather_kernel<<<blocks, 1024, 0, stream>>>(x, T, out, ntok);
}